// LinearAttentionWeights_31731218382955
// MI455X (gfx1250) — compile-verified
//
#include <hip/hip_runtime.h>

typedef __attribute__((ext_vector_type(2))) float v2f;
typedef __attribute__((ext_vector_type(4))) float v4f;
typedef __attribute__((ext_vector_type(8))) float v8f;

constexpr int B = 2, L = 2048, H = 16, D = 64;
constexpr int LT = L / 16; // 128 16-wide tiles along L

// ---------------------------------------------------------------------------
// Kernel 1: per-head linear feature maps.
//   qp[b,h,l,e] = sum_d Q[b,l,h,d] * Wq[h,d,e]   (and same for k)
// One wave per 16-row tile; D=64 -> 16 K-steps of V_WMMA_F32_16X16X4_F32,
// 4 output column tiles (e in 4 groups of 16).
// grid.x = 2 * B*H*(L/16) = 8192, block = 32 (one wave)
// ---------------------------------------------------------------------------
__global__ __launch_bounds__(32) void lfm_project_kernel(
    const float* __restrict__ q, const float* __restrict__ k,
    const float* __restrict__ Wq, const float* __restrict__ Wk,
    float* __restrict__ qp, float* __restrict__ kp)
{
    int idx    = blockIdx.x;
    int tensor = idx >> 12;          // 0 = q, 1 = k (4096 tiles each)
    idx &= 4095;

    const float* x   = tensor ? k  : q;
    const float* W   = tensor ? Wk : Wq;
    float*       out = tensor ? kp : qp;

    int bh = idx >> 7;               // (b*H + h)
    int rt = idx & 127;              // row tile
    int b  = bh >> 4;
    int h  = bh & 15;

    int lane = threadIdx.x;
    int half = lane >> 4;            // 0: lanes 0-15, 1: lanes 16-31
    int lid  = lane & 15;
    int kb   = half * 2;             // K offset within a 4-step: {0,1} or {2,3}

    // A fragments: 16 rows of x for this tile, all 16 K-steps kept in VGPRs.
    v2f a[16];
    const float* arow = x + ((size_t)(b * L + rt * 16 + lid) * H + h) * D;
#pragma unroll
    for (int kk = 0; kk < 16; ++kk)
        a[kk] = *(const v2f*)(arow + kk * 4 + kb);

    const float* wbase = W + (size_t)h * D * D;

#pragma unroll
    for (int et = 0; et < 4; ++et) {
        v8f c = {};
#pragma unroll
        for (int kk = 0; kk < 16; ++kk) {
            int kd = kk * 4 + kb;
            v2f bf;
            bf.x = wbase[(size_t)kd * D + et * 16 + lid];
            bf.y = wbase[(size_t)(kd + 1) * D + et * 16 + lid];
            c = __builtin_amdgcn_wmma_f32_16x16x4_f32(
                    false, a[kk], false, bf, (short)0, c, false, false);
        }
        // C layout: VGPR i -> row M = i + 8*half, col N = lid
        // qp/kp are re-read by the attention pass: keep stores regular-temporal.
        float* obase = out + ((size_t)bh * L + rt * 16) * D + et * 16 + lid;
#pragma unroll
        for (int i = 0; i < 8; ++i) {
            int M = i + half * 8;
            obase[(size_t)M * D] = c[i];
        }
    }
}

// ---------------------------------------------------------------------------
// Kernel 2: attn[b,h,l,s] = (qp[b,h,l,:] . kp[b,h,s,:]) / rowsum
// One workgroup = 8 waves handles 16 rows x full 2048 columns of one (b,h).
// The unnormalized strip is staged in 128 KB LDS (CDNA5: up to 320 KB/WG),
// row sums are reduced from LDS, then the normalized strip streams to HBM
// with NON-TEMPORAL stores so the 536 MB result does not evict the 32 MB
// qp/kp working set from the 192 MB L2.
// grid.x = B*H*(L/16) = 4096, block = 256 (8 waves)
// ---------------------------------------------------------------------------
__global__ __launch_bounds__(256) void attn_norm_kernel(
    const float* __restrict__ qp, const float* __restrict__ kp,
    float* __restrict__ out)
{
    __shared__ float s_attn[16 * L];   // 128 KB strip: 16 rows x 2048 cols
    __shared__ float s_part[16][17];   // per-row partial sums (+pad)
    __shared__ float s_rinv[16];       // reciprocal row sums

    int wg = blockIdx.x;
    int bh = wg >> 7;
    int rt = wg & 127;

    int tid  = threadIdx.x;
    int wid  = tid >> 5;
    int lane = tid & 31;
    int half = lane >> 4;
    int lid  = lane & 15;
    int kb   = half * 2;

    // A fragments: the workgroup's 16 q-rows (same for every wave).
    v2f a[16];
    const float* arow = qp + ((size_t)bh * L + rt * 16 + lid) * D;
#pragma unroll
    for (int kk = 0; kk < 16; ++kk)
        a[kk] = *(const v2f*)(arow + kk * 4 + kb);

    // Each wave owns column tiles ct = wid, wid+8, ... (16 tiles/wave).
    for (int ct = wid; ct < LT; ct += 8) {
        const float* brow = kp + ((size_t)bh * L + ct * 16 + lid) * D + kb;
        v8f c = {};
#pragma unroll
        for (int kk = 0; kk < 16; ++kk) {
            v2f bf = *(const v2f*)(brow + kk * 4);
            c = __builtin_amdgcn_wmma_f32_16x16x4_f32(
                    false, a[kk], false, bf, (short)0, c, false, false);
        }
#pragma unroll
        for (int i = 0; i < 8; ++i) {
            int M = i + half * 8;
            s_attn[M * L + ct * 16 + lid] = c[i];
        }
    }
    __syncthreads();

    // Row sums: 16 threads per row, each sums a 128-wide segment.
    {
        int row = tid >> 4;
        int seg = tid & 15;
        const float* rp = s_attn + row * L + seg * 128;
        float s = 0.f;
#pragma unroll 8
        for (int j = 0; j < 128; ++j) s += rp[j];
        s_part[row][seg] = s;
    }
    __syncthreads();
    if (tid < 16) {
        float s = 0.f;
#pragma unroll
        for (int j = 0; j < 16; ++j) s += s_part[tid][j];
        s_rinv[tid] = 1.0f / s;
    }
    __syncthreads();

    // Normalize + stream out (float4, lane-interleaved, non-temporal).
    {
        int row = tid >> 4;
        int seg = tid & 15;
        float inv = s_rinv[row];
        const v4f* src = (const v4f*)(s_attn + row * L);
        v4f* dst = (v4f*)(out + ((size_t)bh * L + rt * 16 + row) * (size_t)L);
#pragma unroll 4
        for (int j = 0; j < 32; ++j) {
            int c4 = seg + 16 * j;
            v4f v = src[c4];
            v.x *= inv; v.y *= inv; v.z *= inv; v.w *= inv;
            __builtin_nontemporal_store(v, dst + c4);
        }
    }
}

// ---------------------------------------------------------------------------
extern "C" void kernel_launch(void* const* d_in, const int* in_sizes, int n_in,
                              void* d_out, int out_size, void* d_ws, size_t ws_size,
                              hipStream_t stream) {
    const float* q  = (const float*)d_in[0];  // [B,L,H,D]
    const float* k  = (const float*)d_in[1];  // [B,L,H,D]
    const float* Wq = (const float*)d_in[2];  // [H,D,D]
    const float* Wk = (const float*)d_in[3];  // [H,D,D]
    float* out = (float*)d_out;               // [B,H,L,L]

    float* qp = (float*)d_ws;                         // [B,H,L,D] 16 MB
    float* kp = qp + (size_t)B * H * L * D;           // [B,H,L,D] 16 MB

    lfm_project_kernel<<<2 * B * H * (L / 16), 32, 0, stream>>>(q, k, Wq, Wk, qp, kp);
    attn_norm_kernel<<<B * H * (L / 16), 256, 0, stream>>>(qp, kp, out);
}